// KnowledgeGraphEmbedding_13082470383774
// MI455X (gfx1250) — compile-verified
//
#include <hip/hip_runtime.h>
#include <hip/hip_bf16.h>

// Problem constants (match reference)
#define P_REL 1024
#define H_DIM 300
#define E_DIM 300
#define KT    100     // k-extent staged in LDS per wave (300 = 3 * 100)
#define NWAVES 8
#define WAVE  32

typedef float v2f __attribute__((ext_vector_type(2)));
typedef float v8f __attribute__((ext_vector_type(8)));

// One workgroup per relation p. 8 waves, each wave owns H-tiles of 16 rows
// (tiles wave, wave+8, ... of ceil(300/16)=19). Per tile, wave stages 16xKT
// f32 weight blocks into a wave-private LDS buffer with coalesced b128 loads,
// then feeds V_WMMA_F32_16X16X4_F32 with:
//   A[16x4]: lane m = lane&15, VGPR0/1 hold K = koff, koff+1 (koff = lane<16?0:2)
//   B[4x16]: query vector chunk broadcast along N (B[k,n] = w[k]); per ISA-
//            symmetric layout lane holds rows K = koff, koff+1 -> b = {w[k+koff], w[k+koff+1]}
// Accumulating L-tiles with wi then R-tiles with (-wj) leaves diff = li - rj
// in the accumulator (every D column identical). Sum of squares of the 8
// accumulator regs + shuffle over lane^16 gives the 16-row tile's distance.
__global__ __launch_bounds__(256)
void kge_wmma_kernel(const float* __restrict__ tag_rep,
                     const float* __restrict__ Lp,
                     const float* __restrict__ Rp,
                     const int*   __restrict__ relation,
                     const int*   __restrict__ t1p,
                     const int*   __restrict__ t2p,
                     float*       __restrict__ out)
{
    __shared__ float wbuf[NWAVES * 16 * KT];  // 12800 f32 = 51200 B, wave-private slices
    __shared__ float wi[E_DIM];               // tag_rep[tag1]
    __shared__ float wjn[E_DIM];              // -tag_rep[tag2]
    __shared__ float dist_acc;

    const int p    = blockIdx.x;
    const int tid  = threadIdx.x;
    const int lane = tid & (WAVE - 1);
    const int wave = tid >> 5;

    // Stage query vectors (wj pre-negated so one accumulator computes li - rj).
    const int t1 = t1p[0];
    const int t2 = t2p[0];
    for (int e = tid; e < E_DIM; e += 256) {
        wi[e]  =  tag_rep[(size_t)t1 * E_DIM + e];
        wjn[e] = -tag_rep[(size_t)t2 * E_DIM + e];
    }
    if (tid == 0) dist_acc = 0.0f;
    __syncthreads();

    float* mybuf    = &wbuf[wave * 16 * KT];
    const int rsel  = lane & 15;             // A-fragment row M
    const int koff  = (lane >> 4) << 1;      // 0 for lanes 0-15, 2 for lanes 16-31

    float wsum = 0.0f;                       // this wave's partial distance

    for (int tile = wave; tile * 16 < H_DIM; tile += NWAVES) {
        const int h0 = tile * 16;
        v8f c = {0.f, 0.f, 0.f, 0.f, 0.f, 0.f, 0.f, 0.f};

        for (int mat = 0; mat < 2; ++mat) {
            const float* gbase = ((mat == 0) ? Lp : Rp) + (size_t)p * H_DIM * E_DIM;
            const float* wvec  = (mat == 0) ? wi : wjn;   // LDS (addrspace inferred)

            for (int ks = 0; ks < E_DIM; ks += KT) {
                // ---- stage 16 x KT block, coalesced float4, pad rows -> 0 ----
                for (int idx = lane; idx < 16 * (KT / 4); idx += WAVE) {
                    const int row = idx / (KT / 4);
                    const int col = (idx % (KT / 4)) * 4;
                    const int h   = h0 + row;
                    float4 v = make_float4(0.f, 0.f, 0.f, 0.f);
                    if (h < H_DIM) {
                        const float* gp = gbase + (size_t)h * E_DIM + ks + col;
                        v = *(const float4*)gp;           // 16B aligned by construction
                        __builtin_prefetch(gp + KT, 0, 0); // speculative next-stage prefetch
                    }
                    *(float4*)(mybuf + row * KT + col) = v;
                }
                // LDS ops from one wave are in-order (DScnt): stores visible to reads.

                // ---- 25 WMMA accumulation steps over this k-stage ----
                for (int kc = 0; kc < KT; kc += 4) {
                    v2f a, b;
                    const float* ap = mybuf + rsel * KT + kc + koff;  // conflict-free b64
                    a.x = ap[0];
                    a.y = ap[1];
                    const float* bp = wvec + ks + kc + koff;          // broadcast read
                    b.x = bp[0];
                    b.y = bp[1];
                    c = __builtin_amdgcn_wmma_f32_16x16x4_f32(
                            false, a, false, b, (short)0, c, false, false);
                }
            }
        }

        // c[g] = diff[m] (broadcast over all 16 columns).
        // lanes 0-15 hold rows 0-7, lanes 16-31 hold rows 8-15.
        float s = 0.f;
#pragma unroll
        for (int g = 0; g < 8; ++g) s += c[g] * c[g];
        wsum += s;
    }

    // Pair lane l with lane l^16 -> full tile sums (all columns identical).
    wsum += __shfl_xor(wsum, 16, WAVE);
    if (lane == 0) atomicAdd(&dist_acc, wsum);   // ds_add_f32
    __syncthreads();

    if (tid == 0) {
        const float dist = dist_acc;
        const int rel = relation[p];
        out[0 * P_REL + p] = (rel == 1) ? dist : 0.f;
        out[1 * P_REL + p] = (rel == 0) ? dist : 0.f;
        out[2 * P_REL + p] = (rel == 1) ? 1.f : 0.f;
        out[3 * P_REL + p] = (rel == 0) ? 1.f : 0.f;
    }
}

extern "C" void kernel_launch(void* const* d_in, const int* in_sizes, int n_in,
                              void* d_out, int out_size, void* d_ws, size_t ws_size,
                              hipStream_t stream) {
    const float* tag_rep  = (const float*)d_in[0];
    const float* Lp       = (const float*)d_in[1];
    const float* Rp       = (const float*)d_in[2];
    const int*   relation = (const int*)d_in[3];
    const int*   t1       = (const int*)d_in[4];
    const int*   t2       = (const int*)d_in[5];
    float*       out      = (float*)d_out;
    (void)in_sizes; (void)n_in; (void)out_size; (void)d_ws; (void)ws_size;

    kge_wmma_kernel<<<dim3(P_REL), dim3(256), 0, stream>>>(
        tag_rep, Lp, Rp, relation, t1, t2, out);
}